// ManifoldHyperConnectionFuse_39977555591475
// MI455X (gfx1250) — compile-verified
//
#include <hip/hip_runtime.h>
#include <math.h>

// ManifoldHyperConnection fused kernel for gfx1250 (MI455X).
// B=2, L=2048, N=4, D=2048 -> BL=4096 tokens, nc=8192, nout=24.

typedef __attribute__((ext_vector_type(2))) float v2f;
typedef __attribute__((ext_vector_type(4))) float v4f;
typedef __attribute__((ext_vector_type(8))) float v8f;

#define TPW    16      // tokens per workgroup (WMMA M)
#define NDIM   4       // N
#define DDIM   2048    // D
#define NC     8192    // N*D
#define NOUT   24      // N*N + 2N
#define CHUNK  256     // K chunk staged in LDS
#define HPAD   260     // padded row stride for h tile (bank-conflict free)
#define WTROW  260     // padded row stride for transposed w tile
#define KPW    32      // K elements per wave per chunk (8 waves * 32 = 256)

__global__ __launch_bounds__(256)
void mhc_fused_kernel(const float* __restrict__ h,
                      const float* __restrict__ gamma,
                      const float* __restrict__ w,
                      const float* __restrict__ alpha,
                      const float* __restrict__ beta,
                      const int*   __restrict__ sk_it,
                      float* __restrict__ out)
{
    __shared__ float lds_h[TPW * HPAD];      // 16.6 KB: (gamma*h) chunk, 16 tokens x 256 K
    __shared__ float lds_wT[32 * WTROW];     // 33.3 KB: w chunk TRANSPOSED, 32 cols x 256 K
    __shared__ float lds_H[TPW * 32];        // reduced H = hvec @ w  (cols 24..31 zero)
    __shared__ float lds_ss[TPW];            // per-token sum of squares (raw h)
    __shared__ float lds_Q[TPW * 16];        // fused 4x4 mixing matrix per token

    const int t     = threadIdx.x;
    const int wave  = t >> 5;
    const int lane  = t & 31;
    const int half  = lane >> 4;       // 0: lanes 0-15, 1: lanes 16-31
    const int mrow  = lane & 15;       // M row (A) / N col (B,C)
    const int ssrow = t >> 6;          // 0..3, fixed row-group for sumsq partials
    const int col4  = t & 63;          // chunk-invariant staging column (float4 units)
    const int tokBase = blockIdx.x * TPW;

    const float* __restrict__ hWG = h + (size_t)tokBase * NC;

    // ---- one-time LDS init: H accumulator, sumsq, wT pad rows (cols 24..31) ----
    for (int i = t; i < TPW * 32; i += 256) lds_H[i] = 0.0f;
    if (t < TPW) lds_ss[t] = 0.0f;
    for (int i = t; i < 8 * WTROW; i += 256) lds_wT[NOUT * WTROW + i] = 0.0f;

    v8f c0 = {};   // outputs 0..15
    v8f c1 = {};   // outputs 16..23 (24..31 padded zero)
    float ss0 = 0.f, ss1 = 0.f, ss2 = 0.f, ss3 = 0.f;

    // =================== Phase 1: H = (gamma*h) @ w via WMMA f32 ===================
    for (int k0 = 0; k0 < NC; k0 += CHUNK) {
        __syncthreads();   // protect LDS tiles from previous iteration's readers

        // gamma for this thread's staging column (shared by all 4 rows it stages)
        const v4f g4 = *(const v4f*)(gamma + k0 + col4 * 4);

        // stage w chunk transposed: flat coalesced global read, scattered LDS write
        {
            const float* __restrict__ wsrc = w + (size_t)k0 * NOUT;
            for (int i = t; i < CHUNK * NOUT; i += 256) {
                float val = wsrc[i];
                int row = i / NOUT;            // K index within chunk
                int col = i - row * NOUT;      // output index
                lds_wT[col * WTROW + row] = val;
            }
        }

        // stage h chunk pre-scaled by gamma: 16 rows x 64 float4, 4 rows per thread
        #pragma unroll
        for (int i = 0; i < 4; ++i) {
            const int row = i * 4 + ssrow;
            const v4f v4 = *(const v4f*)(hWG + (size_t)row * NC + k0 + col4 * 4);
            float s = v4.x * v4.x + v4.y * v4.y + v4.z * v4.z + v4.w * v4.w;
            if      (i == 0) ss0 += s;
            else if (i == 1) ss1 += s;
            else if (i == 2) ss2 += s;
            else             ss3 += s;
            float* dst = &lds_h[row * HPAD + col4 * 4];
            dst[0] = v4.x * g4.x; dst[1] = v4.y * g4.y;
            dst[2] = v4.z * g4.z; dst[3] = v4.w * g4.w;
            if (k0 + CHUNK < NC)  // prefetch next chunk (global_prefetch_b8)
                __builtin_prefetch(hWG + (size_t)row * NC + (k0 + CHUNK) + col4 * 4, 0, 0);
        }
        __syncthreads();

        // each wave: its own 32-wide K slice, 8 WMMA steps of K=4, two N tiles
        const int kw = wave * KPW;
        #pragma unroll
        for (int s = 0; s < KPW / 4; ++s) {
            const int kk = kw + s * 4 + half * 2;   // chunk-local K of this lane-half
            const v2f a  = *(const v2f*)(&lds_h[mrow * HPAD + kk]);
            const v2f b0 = *(const v2f*)(&lds_wT[mrow * WTROW + kk]);
            const v2f b1 = *(const v2f*)(&lds_wT[(16 + mrow) * WTROW + kk]);
            c0 = __builtin_amdgcn_wmma_f32_16x16x4_f32(
                     false, a, false, b0, (short)0, c0, false, false);
            c1 = __builtin_amdgcn_wmma_f32_16x16x4_f32(
                     false, a, false, b1, (short)0, c1, false, false);
        }
    }

    // cross-wave reduction of WMMA accumulators + sumsq into LDS
    __syncthreads();
    #pragma unroll
    for (int r = 0; r < 8; ++r) {
        const int m = r + half * 8;    // C layout: lanes 0-15 M=0..7, lanes 16-31 M=8..15
        atomicAdd(&lds_H[m * 32 + mrow],      c0[r]);
        atomicAdd(&lds_H[m * 32 + 16 + mrow], c1[r]);
    }
    atomicAdd(&lds_ss[ssrow],      ss0);
    atomicAdd(&lds_ss[4 + ssrow],  ss1);
    atomicAdd(&lds_ss[8 + ssrow],  ss2);
    atomicAdd(&lds_ss[12 + ssrow], ss3);
    __syncthreads();

    // =================== Phase 2: gates + exp + Sinkhorn -> Q (16 threads) ===================
    if (t < TPW) {
        const int iters = sk_it[0];
        const float r_ = sqrtf((float)NC / (lds_ss[t] + 1e-30f));  // 1/r
        const float a0 = alpha[0], a1 = alpha[1], a2 = alpha[2];

        float pre[NDIM], post[NDIM], Mm[NDIM * NDIM];
        #pragma unroll
        for (int n = 0; n < NDIM; ++n) {
            float xp = r_ * lds_H[t * 32 + n] * a0 + beta[n];
            pre[n]  = 1.0f / (1.0f + expf(-xp));
            float xq = r_ * lds_H[t * 32 + NDIM + n] * a1 + beta[NDIM + n];
            post[n] = 2.0f / (1.0f + expf(-xq));
        }
        #pragma unroll
        for (int i = 0; i < NDIM * NDIM; ++i)
            Mm[i] = expf(r_ * lds_H[t * 32 + 2 * NDIM + i] * a2 + beta[2 * NDIM + i]);

        float u[NDIM] = {1.f, 1.f, 1.f, 1.f};
        float v[NDIM] = {1.f, 1.f, 1.f, 1.f};
        for (int it = 0; it < iters; ++it) {
            #pragma unroll
            for (int i = 0; i < NDIM; ++i) {
                float s = 1e-8f;
                #pragma unroll
                for (int j = 0; j < NDIM; ++j) s += Mm[i * NDIM + j] * v[j];
                u[i] = 1.0f / s;
            }
            #pragma unroll
            for (int j = 0; j < NDIM; ++j) {
                float s = 1e-8f;
                #pragma unroll
                for (int i = 0; i < NDIM; ++i) s += Mm[i * NDIM + j] * u[i];
                v[j] = 1.0f / s;
            }
        }
        // Q[m][n] = H_post[m]*H_pre[n] + P[m][n],  P = diag(u) M diag(v)
        #pragma unroll
        for (int m = 0; m < NDIM; ++m)
            #pragma unroll
            for (int n = 0; n < NDIM; ++n)
                lds_Q[t * 16 + m * NDIM + n] =
                    post[m] * pre[n] + u[m] * Mm[m * NDIM + n] * v[n];
    }
    __syncthreads();

    // ========= Phase 3: out = Q @ h (per token; NT loads/stores, h is dead after) =========
    float* __restrict__ outWG = out + (size_t)tokBase * NC;
    for (int tok = 0; tok < TPW; ++tok) {
        const float* __restrict__ hrow = hWG   + (size_t)tok * NC;
        float*       __restrict__ orow = outWG + (size_t)tok * NC;
        float q[16];
        #pragma unroll
        for (int i = 0; i < 16; ++i) q[i] = lds_Q[tok * 16 + i];  // LDS broadcast

        #pragma unroll
        for (int it2 = 0; it2 < 2; ++it2) {
            const int d4 = (it2 * 256 + t) * 4;   // 512 float4 columns / 256 threads
            const v4f h0 = __builtin_nontemporal_load((const v4f*)(hrow + 0 * DDIM + d4));
            const v4f h1 = __builtin_nontemporal_load((const v4f*)(hrow + 1 * DDIM + d4));
            const v4f h2 = __builtin_nontemporal_load((const v4f*)(hrow + 2 * DDIM + d4));
            const v4f h3 = __builtin_nontemporal_load((const v4f*)(hrow + 3 * DDIM + d4));
            #pragma unroll
            for (int m = 0; m < NDIM; ++m) {
                v4f o;
                o.x = q[m*4+0]*h0.x + q[m*4+1]*h1.x + q[m*4+2]*h2.x + q[m*4+3]*h3.x;
                o.y = q[m*4+0]*h0.y + q[m*4+1]*h1.y + q[m*4+2]*h2.y + q[m*4+3]*h3.y;
                o.z = q[m*4+0]*h0.z + q[m*4+1]*h1.z + q[m*4+2]*h2.z + q[m*4+3]*h3.z;
                o.w = q[m*4+0]*h0.w + q[m*4+1]*h1.w + q[m*4+2]*h2.w + q[m*4+3]*h3.w;
                __builtin_nontemporal_store(o, (v4f*)(orow + m * DDIM + d4));
            }
        }
    }
}

extern "C" void kernel_launch(void* const* d_in, const int* in_sizes, int n_in,
                              void* d_out, int out_size, void* d_ws, size_t ws_size,
                              hipStream_t stream) {
    const float* h     = (const float*)d_in[0];
    const float* gamma = (const float*)d_in[1];
    const float* w     = (const float*)d_in[2];
    const float* alpha = (const float*)d_in[3];
    const float* beta  = (const float*)d_in[4];
    const int*   skit  = (const int*)d_in[5];
    float* out = (float*)d_out;

    const int BL = in_sizes[0] / NC;     // 4096 tokens
    const int nblocks = BL / TPW;        // 256 workgroups

    mhc_fused_kernel<<<nblocks, 256, 0, stream>>>(h, gamma, w, alpha, beta, skit, out);
}